// NHGrid_74440373175053
// MI455X (gfx1250) — compile-verified
//
#include <hip/hip_runtime.h>

typedef __attribute__((ext_vector_type(2))) float v2f;
typedef __attribute__((ext_vector_type(8))) float v8f;

__device__ __forceinline__ float lrelu(float v) { return v > 0.0f ? v : 0.01f * v; }

// spacing_i = 256.0 // 1.6**(11-i), precomputed exactly
static __device__ const float c_spacing[12] = {
    1.f, 2.f, 3.f, 5.f, 9.f, 15.f, 24.f, 39.f, 62.f, 100.f, 160.f, 256.f
};

#define WMMA_F32(acc, a, b) \
    __builtin_amdgcn_wmma_f32_16x16x4_f32(false, (a), false, (b), (short)0, (acc), false, false)

// LDS strides (floats), chosen even (8B-aligned b64 access) and conflict-free:
// feat 16x26, act1 16x34, act2 16x18, act3 16x10 (act2 reuses feat, act3 reuses act1)
#define FEAT_STRIDE 26
#define A1_STRIDE   34
#define A2_STRIDE   18
#define A3_STRIDE   10
#define BUF0_SZ     (16 * FEAT_STRIDE)   // 416
#define BUF1_SZ     (16 * A1_STRIDE)     // 544

__global__ __launch_bounds__(256) void ngp_hashmlp_kernel(
    const float* __restrict__ xg, const float* __restrict__ tables,
    const float* __restrict__ W1, const float* __restrict__ b1,
    const float* __restrict__ W2, const float* __restrict__ b2,
    const float* __restrict__ W3, const float* __restrict__ b3,
    const float* __restrict__ W4, const float* __restrict__ b4,
    float* __restrict__ out, int npts)
{
    __shared__ __align__(16) float sW1[32 * 26];  // W1[n][k], k<24, padded stride 26
    __shared__ __align__(16) float sB1[32];
    __shared__ __align__(16) float sW2[16 * 34];  // W2[n][k], k<32, stride 34
    __shared__ __align__(16) float sB2[16];
    __shared__ __align__(16) float sW3[16 * 18];  // W3 padded to 16 rows (8..15 zero), stride 18
    __shared__ __align__(16) float sB3[16];
    __shared__ __align__(16) float sbuf0[8 * BUF0_SZ];
    __shared__ __align__(16) float sbuf1[8 * BUF1_SZ];

    const int tid = threadIdx.x;

    // ---- stage weights into LDS (once per block) ----
    for (int i = tid; i < 32 * 24; i += 256) { int r = i / 24, c = i - r * 24; sW1[r * 26 + c] = W1[i]; }
    for (int i = tid; i < 16 * 32; i += 256) { int r = i >> 5, c = i & 31;      sW2[r * 34 + c] = W2[i]; }
    for (int i = tid; i < 16 * 16; i += 256) { int r = i >> 4, c = i & 15;      sW3[r * 18 + c] = (r < 8) ? W3[r * 16 + c] : 0.0f; }
    if (tid < 32) sB1[tid] = b1[tid];
    if (tid < 16) sB2[tid] = b2[tid];
    if (tid < 16) sB3[tid] = (tid < 8) ? b3[tid] : 0.0f;
    __syncthreads();

    const int lane = tid & 31;
    const int wave = tid >> 5;
    float* feat = &sbuf0[wave * BUF0_SZ];   // 16 x 26, later reused as act2 16 x 18
    float* act1 = &sbuf1[wave * BUF1_SZ];   // 16 x 34, later reused as act3 16 x 10

    const int base = blockIdx.x * 128 + wave * 16;

    // ---- hash + gather: 12 levels x 16 points = 192 tasks over 32 lanes ----
    #pragma unroll
    for (int it = 0; it < 6; ++it) {
        int task = it * 32 + lane;
        int lvl = task >> 4;       // 2 levels per iteration
        int p   = task & 15;
        int pid = base + p; if (pid >= npts) pid = npts - 1;
        v2f xv = *(const v2f*)(xg + (size_t)pid * 2);
        float s  = c_spacing[lvl];
        float x0 = xv.x * 0.5f + 0.5f;
        float x1 = xv.y * 0.5f + 0.5f;
        int d0 = (int)floorf(x0 * s);
        int d1 = (int)floorf(x1 * s);
        // int64 hash mod 2^19 == low 19 bits of 32-bit wrapping arithmetic
        unsigned h = ((unsigned)d0 * 73856093u + (unsigned)d1 * 19349663u) & 0x7FFFFu;
        v2f f = *(const v2f*)(tables + (((size_t)lvl << 19) + h) * 2);
        *(v2f*)&feat[p * FEAT_STRIDE + lvl * 2] = f;   // even offset -> aligned
    }

    const int m      = lane & 15;          // A row / B col within tile
    const int kh     = (lane >> 4) << 1;   // K-half offset: 0 (lanes 0-15) or 2 (16-31)
    const int rowoff = (lane >> 4) << 3;   // C/D row offset: 0 or 8

    // ---- layer 1: [16x24] @ [24x32] -> two 16x16 WMMA tiles, K in 6 steps of 4 ----
    v8f acc0 = {}; v8f acc1 = {};
    #pragma unroll
    for (int kk = 0; kk < 6; ++kk) {
        int klo = kk * 4 + kh;
        v2f a  = *(const v2f*)&feat[m * FEAT_STRIDE + klo];
        v2f bA = *(const v2f*)&sW1[m * 26 + klo];          // out cols 0..15
        v2f bB = *(const v2f*)&sW1[(16 + m) * 26 + klo];   // out cols 16..31
        acc0 = WMMA_F32(acc0, a, bA);
        acc1 = WMMA_F32(acc1, a, bB);
    }
    #pragma unroll
    for (int r = 0; r < 8; ++r) {
        int mm = r + rowoff;
        act1[mm * A1_STRIDE + m]      = lrelu(acc0[r] + sB1[m]);
        act1[mm * A1_STRIDE + 16 + m] = lrelu(acc1[r] + sB1[16 + m]);
    }

    // ---- layer 2: [16x32] @ [32x16], K in 8 steps of 4 ----
    v8f acc2 = {};
    #pragma unroll
    for (int kk = 0; kk < 8; ++kk) {
        int klo = kk * 4 + kh;
        v2f a = *(const v2f*)&act1[m * A1_STRIDE + klo];
        v2f b = *(const v2f*)&sW2[m * 34 + klo];
        acc2 = WMMA_F32(acc2, a, b);
    }
    float* act2 = feat;  // reuse buffer, 16 x 18
    #pragma unroll
    for (int r = 0; r < 8; ++r) {
        int mm = r + rowoff;
        act2[mm * A2_STRIDE + m] = lrelu(acc2[r] + sB2[m]);
    }

    // ---- layer 3: [16x16] @ [16x8 padded to 16], K in 4 steps of 4 ----
    v8f acc3 = {};
    #pragma unroll
    for (int kk = 0; kk < 4; ++kk) {
        int klo = kk * 4 + kh;
        v2f a = *(const v2f*)&act2[m * A2_STRIDE + klo];
        v2f b = *(const v2f*)&sW3[m * 18 + klo];
        acc3 = WMMA_F32(acc3, a, b);
    }
    float* act3 = act1;  // reuse buffer, 16 x 10
    if (m < 8) {
        #pragma unroll
        for (int r = 0; r < 8; ++r) {
            int mm = r + rowoff;
            act3[mm * A3_STRIDE + m] = lrelu(acc3[r] + sB3[m]);
        }
    }

    // ---- layer 4: [16x8] @ [8x1] + double LeakyReLU, scalar (16 FLOPs/pt) ----
    if (lane < 16) {
        int pid = base + lane;
        float y = b4[0];
        #pragma unroll
        for (int k = 0; k < 8; ++k) y += act3[lane * A3_STRIDE + k] * W4[k];
        y = lrelu(lrelu(y));
        if (pid < npts) out[pid] = y;
    }
}

extern "C" void kernel_launch(void* const* d_in, const int* in_sizes, int n_in,
                              void* d_out, int out_size, void* d_ws, size_t ws_size,
                              hipStream_t stream) {
    (void)n_in; (void)out_size; (void)d_ws; (void)ws_size;
    const float* x      = (const float*)d_in[0];
    const float* tables = (const float*)d_in[1];
    const float* W1 = (const float*)d_in[2]; const float* b1 = (const float*)d_in[3];
    const float* W2 = (const float*)d_in[4]; const float* b2 = (const float*)d_in[5];
    const float* W3 = (const float*)d_in[6]; const float* b3 = (const float*)d_in[7];
    const float* W4 = (const float*)d_in[8]; const float* b4 = (const float*)d_in[9];
    int npts = in_sizes[0] / 2;
    int nblocks = (npts + 127) / 128;   // 128 points per block (8 waves x 16 points)
    ngp_hashmlp_kernel<<<nblocks, 256, 0, stream>>>(
        x, tables, W1, b1, W2, b2, W3, b3, W4, b4, (float*)d_out, npts);
}